// WeightQuantizer_34359738368025
// MI455X (gfx1250) — compile-verified
//
#include <hip/hip_runtime.h>

typedef float v4f __attribute__((ext_vector_type(4)));

#define BLOCK 256
#define GRID  8192

// ---- helpers ------------------------------------------------------------

// bf16 round-to-nearest-even, result back in fp32 (matches jnp bf16 cast)
__device__ __forceinline__ float bf16_rne(float x) {
  unsigned u = __float_as_uint(x);
  unsigned r = (u + 0x7fffu + ((u >> 16) & 1u)) & 0xffff0000u;
  return __uint_as_float(r);
}

// e2m1 quantize of y (already divided by block scale), then re-scale.
// step = 2^(clip(floor(log2|y|),0,2) - 1) -> thresholds at 2.0 and 4.0.
__device__ __forceinline__ float fp4_q_scaled(float y, float scale) {
  float ay = __builtin_fabsf(y);
  float invstep = ay < 2.0f ? 2.0f : (ay < 4.0f ? 1.0f : 0.5f);
  float step    = ay < 2.0f ? 0.5f : (ay < 4.0f ? 1.0f : 2.0f);
  float q = __builtin_rintf(y * invstep) * step;       // v_rndne_f32 (RNE)
  q = fminf(6.0f, fmaxf(-6.0f, q));                    // saturate to +-6
  return q * scale;
}

// async DMA: 16B per lane, global -> LDS (ASYNCcnt path, CDNA5).
// Input is consumed exactly once -> non-temporal load hint.
__device__ __forceinline__ void async_load16(unsigned lds_off, const v4f* gptr) {
  asm volatile("global_load_async_to_lds_b128 %0, %1, off th:TH_LOAD_NT"
               :: "v"(lds_off), "v"(gptr) : "memory");
}

// ---- kernel -------------------------------------------------------------

__global__ __launch_bounds__(BLOCK) void mxfp4_ste_kernel(
    const v4f* __restrict__ in, v4f* __restrict__ out, int n4) {
  __shared__ v4f stage[2][BLOCK];
  const int tid = threadIdx.x;
  const int stride = gridDim.x * BLOCK;         // in float4 units
  int i = blockIdx.x * BLOCK + tid;

  // low 32 bits of a flat LDS pointer are the wave-relative LDS byte offset
  const unsigned loff0 = (unsigned)(unsigned long long)&stage[0][tid];
  const unsigned loff1 = (unsigned)(unsigned long long)&stage[1][tid];

  int buf = 0;
  if (i < n4) async_load16(loff0, in + i);

  for (; i < n4; i += stride) {
    const int nx = i + stride;
    const bool have_next = nx < n4;

    // pull the tile 2 strides ahead toward L2/WGP$ (global_prefetch_b8,
    // counter-free) so the async DMA's final hop is a cache hit and the
    // 2-deep LDS pipeline is enough in-flight data.
    const int pf = i + 2 * stride;
    if (pf < n4) __builtin_prefetch((const void*)(in + pf), 0, 0);

    if (have_next) async_load16(buf ? loff0 : loff1, in + nx);

    // wait until the tile for THIS iteration has landed in LDS
    if (have_next) asm volatile("s_wait_asynccnt 0x1" ::: "memory");
    else           asm volatile("s_wait_asynccnt 0x0" ::: "memory");

    v4f v = stage[buf][tid];                    // ds_load_b128

    // bf16 RNE pre-quantization (reference quantizes before MX step)
    float b0 = bf16_rne(v.x), b1 = bf16_rne(v.y);
    float b2 = bf16_rne(v.z), b3 = bf16_rne(v.w);

    // block max|.| : 4 elems in-lane, then 8-lane group (32 elems = 1 MX block)
    float m = fmaxf(fmaxf(__builtin_fabsf(b0), __builtin_fabsf(b1)),
                    fmaxf(__builtin_fabsf(b2), __builtin_fabsf(b3)));
    m = fmaxf(m, __shfl_xor(m, 1, 32));
    m = fmaxf(m, __shfl_xor(m, 2, 32));
    m = fmaxf(m, __shfl_xor(m, 4, 32));

    // shared_exp = clip(floor(log2(max==0 ? 1 : max)) - 2, -127, 127)
    unsigned mb = __float_as_uint(m);
    int e = (mb != 0u) ? (int)((mb >> 23) & 0xffu) - 127 : 0;
    int sexp = e - 2;
    if (sexp < -127) sexp = -127;
    float scale = ldexpf(1.0f, sexp);           // v_ldexp_f32
    float inv   = ldexpf(1.0f, -sexp);

    // quantize + straight-through estimator: out = x + (qx - x)
    v4f c;
    c.x = v.x + (fp4_q_scaled(b0 * inv, scale) - v.x);
    c.y = v.y + (fp4_q_scaled(b1 * inv, scale) - v.y);
    c.z = v.z + (fp4_q_scaled(b2 * inv, scale) - v.z);
    c.w = v.w + (fp4_q_scaled(b3 * inv, scale) - v.w);

    __builtin_nontemporal_store(c, out + i);    // global_store_b128 th:NT
    buf ^= 1;
  }
}

// ---- launch -------------------------------------------------------------

extern "C" void kernel_launch(void* const* d_in, const int* in_sizes, int n_in,
                              void* d_out, int out_size, void* d_ws, size_t ws_size,
                              hipStream_t stream) {
  const v4f* x = (const v4f*)d_in[0];   // fp32 input, 8192*8192
  v4f* out = (v4f*)d_out;               // fp32 output
  const int n  = in_sizes[0];
  const int n4 = n >> 2;                // float4 count (n divisible by 4)

  int blocks = GRID;
  int need = (n4 + BLOCK - 1) / BLOCK;
  if (blocks > need) blocks = need;

  hipLaunchKernelGGL(mxfp4_ste_kernel, dim3(blocks), dim3(BLOCK), 0, stream,
                     x, out, n4);
}